// SPA_2508260900994
// MI455X (gfx1250) — compile-verified
//
#include <hip/hip_runtime.h>
#include <hip/hip_bf16.h>

// ---------------- problem constants ----------------
#define BB    8
#define CIN   32
#define CI    16
#define HH    128
#define HW    (HH*HH)          // 16384
#define KS    7
#define STR   4
#define NHT   31
#define LL    961              // NHT*NHT
#define FIN   784              // KS*KS*CI
#define FOUT  196
#define F62   1568             // KS*KS*CIN

typedef __attribute__((ext_vector_type(2))) float v2f;
typedef __attribute__((ext_vector_type(8))) float v8f;

// D = A(16x4) * B(4x16) + C, full f32 via CDNA5 WMMA.
__device__ __forceinline__ v8f wmma4(v2f a, v2f b, v8f c) {
  return __builtin_amdgcn_wmma_f32_16x16x4_f32(
      /*neg_a=*/false, a, /*neg_b=*/false, b,
      /*c_mod=*/(short)0, c, /*reuse_a=*/false, /*reuse_b=*/false);
}

__device__ __forceinline__ int clampi(int r, int hi) { return r > hi ? hi : r; }

// ---------------- conv 3x3 with reflect pad(1) ----------------
__global__ void k_conv3x3_reflect(const float* __restrict__ x,
                                  const float* __restrict__ w,
                                  const float* __restrict__ bias,
                                  float* __restrict__ y,
                                  int Cin, int Cout, int leaky) {
  int idx = blockIdx.x * blockDim.x + threadIdx.x;
  int total = BB * Cout * HW;
  if (idx >= total) return;
  int ww = idx & 127;
  int hh = (idx >> 7) & 127;
  int co = (idx >> 14) % Cout;
  int b  = idx / (Cout * HW);
  float acc = bias ? bias[co] : 0.0f;
  for (int ci = 0; ci < Cin; ++ci) {
    const float* xc = x + ((size_t)b * Cin + ci) * HW;
    const float* wc = w + ((size_t)co * Cin + ci) * 9;
#pragma unroll
    for (int kh = 0; kh < 3; ++kh) {
      int h = hh + kh - 1; h = (h < 0) ? 1 : ((h > 127) ? 126 : h);
#pragma unroll
      for (int kw = 0; kw < 3; ++kw) {
        int q = ww + kw - 1; q = (q < 0) ? 1 : ((q > 127) ? 126 : q);
        acc += xc[h * HH + q] * wc[kh * 3 + kw];
      }
    }
  }
  if (leaky) acc = (acc > 0.0f) ? acc : 0.2f * acc;
  y[idx] = acc;
}

// ---------------- conv 1x1 (+optional leaky, +optional residual) --------
__global__ void k_conv1x1(const float* __restrict__ x,
                          const float* __restrict__ w,
                          const float* __restrict__ bias,
                          const float* __restrict__ residual,
                          float* __restrict__ y,
                          int Cin, int Cout, int leaky) {
  int idx = blockIdx.x * blockDim.x + threadIdx.x;
  int total = BB * Cout * HW;
  if (idx >= total) return;
  int p  = idx & (HW - 1);
  int co = (idx >> 14) % Cout;
  int b  = idx / (Cout * HW);
  float acc = bias ? bias[co] : 0.0f;
  const float* xb = x + (size_t)b * Cin * HW;
  for (int ci = 0; ci < Cin; ++ci)
    acc += xb[(size_t)ci * HW + p] * w[co * Cin + ci];
  if (leaky) acc = (acc > 0.0f) ? acc : 0.2f * acc;
  if (residual) acc += residual[idx];
  y[idx] = acc;
}

// ---------------- fused unfold + FC ----------------
__global__ void k_patch_fc(const float* __restrict__ xin,   // (B,CI,128,128)
                           const float* __restrict__ fcw,   // (FOUT,FIN)
                           const float* __restrict__ fcb,   // (FOUT)
                           float* __restrict__ out) {       // (B,LL,FOUT)
  int idx = blockIdx.x * blockDim.x + threadIdx.x;
  int total = BB * LL * FOUT;
  if (idx >= total) return;
  int fo = idx % FOUT;
  int l  = (idx / FOUT) % LL;
  int b  = idx / (FOUT * LL);
  int lh = l / NHT, lw = l % NHT;
  const float* xb   = xin + (size_t)b * CI * HW + (lh * STR) * HH + lw * STR;
  const float* wrow = fcw + (size_t)fo * FIN;
  float acc = fcb[fo];
  for (int c = 0; c < CI; ++c) {
    const float* xc = xb + (size_t)c * HW;
    const float* wc = wrow + c * (KS * KS);
#pragma unroll
    for (int kh = 0; kh < KS; ++kh)
#pragma unroll
      for (int kw = 0; kw < KS; ++kw)
        acc += xc[kh * HH + kw] * wc[kh * KS + kw];
  }
  out[idx] = acc;
}

// ---------------- materialize p62 = unfold(x3) ----------------
__global__ void k_unfold62(const float* __restrict__ x3,    // (B,CIN,128,128)
                           float* __restrict__ p62) {
  int idx = blockIdx.x * blockDim.x + threadIdx.x;
  int total = BB * LL * F62;
  if (idx >= total) return;
  int fi = idx % F62;
  int l  = (idx / F62) % LL;
  int b  = idx / (F62 * LL);
  int c  = fi / 49, r = fi % 49, kh = r / 7, kw = r % 7;
  int lh = l / NHT, lw = l % NHT;
  p62[idx] = x3[(((size_t)b * CIN + c) * HH + (lh * STR + kh)) * HH + (lw * STR + kw)];
}

// ---------------- Mw = mconv_w + mfc_w ----------------
__global__ void k_addmw(const float* __restrict__ a, const float* __restrict__ b,
                        float* __restrict__ o) {
  int idx = blockIdx.x * blockDim.x + threadIdx.x;
  if (idx < LL * LL) o[idx] = a[idx] + b[idx];
}

// ---------------- WMMA GEMM: A[b] = wf[b](961x196) @ xf[b]^T ----------------
// 4 waves per block; each wave: 64x16 D block (4 named accumulators).
__global__ void k_gemm_score(const float* __restrict__ wf,
                             const float* __restrict__ xf,
                             float* __restrict__ A) {
  int tl = blockIdx.x, b = blockIdx.z;
  int wave = threadIdx.x >> 5, lane = threadIdx.x & 31;
  int tm = blockIdx.y * 4 + wave;                 // 64 col tiles (61 valid)
  int half = lane >> 4, i16 = lane & 15, koff = half * 2;
  const float* wfb = wf + (size_t)b * LL * FOUT;
  const float* xfb = xf + (size_t)b * LL * FOUT;
  const float* arow0 = wfb + (size_t)clampi(tl * 64 +  0 + i16, LL - 1) * FOUT + koff;
  const float* arow1 = wfb + (size_t)clampi(tl * 64 + 16 + i16, LL - 1) * FOUT + koff;
  const float* arow2 = wfb + (size_t)clampi(tl * 64 + 32 + i16, LL - 1) * FOUT + koff;
  const float* arow3 = wfb + (size_t)clampi(tl * 64 + 48 + i16, LL - 1) * FOUT + koff;
  int ncol = tm * 16 + i16;
  const float* brow = xfb + (size_t)clampi(ncol, LL - 1) * FOUT + koff;
  v8f acc0 = {}, acc1 = {}, acc2 = {}, acc3 = {};
#pragma clang loop unroll(disable)
  for (int kk = 0; kk < 192; kk += 16) {
#pragma unroll
    for (int u = 0; u < 4; ++u) {
      int k = kk + u * 4;
      v2f bv = *(const v2f*)(brow + k);           // 16B-aligned row stride
      acc0 = wmma4(*(const v2f*)(arow0 + k), bv, acc0);
      acc1 = wmma4(*(const v2f*)(arow1 + k), bv, acc1);
      acc2 = wmma4(*(const v2f*)(arow2 + k), bv, acc2);
      acc3 = wmma4(*(const v2f*)(arow3 + k), bv, acc3);
    }
  }
  { // remainder: k = 192 (196 = 12*16 + 4)
    v2f bv = *(const v2f*)(brow + 192);
    acc0 = wmma4(*(const v2f*)(arow0 + 192), bv, acc0);
    acc1 = wmma4(*(const v2f*)(arow1 + 192), bv, acc1);
    acc2 = wmma4(*(const v2f*)(arow2 + 192), bv, acc2);
    acc3 = wmma4(*(const v2f*)(arow3 + 192), bv, acc3);
  }
  if (ncol < LL) {
    float* Ab = A + (size_t)b * LL * LL + ncol;
#pragma unroll
    for (int r = 0; r < 8; ++r) {
      int row = tl * 64 + half * 8 + r;
      if (row      < LL) Ab[(size_t)(row     ) * LL] = acc0[r];
      if (row + 16 < LL) Ab[(size_t)(row + 16) * LL] = acc1[r];
      if (row + 32 < LL) Ab[(size_t)(row + 32) * LL] = acc2[r];
      if (row + 48 < LL) Ab[(size_t)(row + 48) * LL] = acc3[r];
    }
  }
}

// ---------------- WMMA GEMM: Ti[b] = Mw(961x961) @ A[b] + bias ----------------
__global__ void k_gemm_ti(const float* __restrict__ Mw,
                          const float* __restrict__ mcb,
                          const float* __restrict__ mfb,
                          const float* __restrict__ A,
                          float* __restrict__ Ti) {
  int tn = blockIdx.x, b = blockIdx.z;
  int wave = threadIdx.x >> 5, lane = threadIdx.x & 31;
  int tm = blockIdx.y * 4 + wave;
  int half = lane >> 4, i16 = lane & 15, koff = half * 2;
  const float* arow0 = Mw + (size_t)clampi(tn * 64 +  0 + i16, LL - 1) * LL + koff;
  const float* arow1 = Mw + (size_t)clampi(tn * 64 + 16 + i16, LL - 1) * LL + koff;
  const float* arow2 = Mw + (size_t)clampi(tn * 64 + 32 + i16, LL - 1) * LL + koff;
  const float* arow3 = Mw + (size_t)clampi(tn * 64 + 48 + i16, LL - 1) * LL + koff;
  int mcol = tm * 16 + i16;
  const float* bcol = A + (size_t)b * LL * LL + clampi(mcol, LL - 1);
  v8f acc0 = {}, acc1 = {}, acc2 = {}, acc3 = {};
#pragma clang loop unroll(disable)
  for (int kk = 0; kk < 960; kk += 16) {          // 60 iterations, branch-free
    __builtin_prefetch(bcol + (size_t)(kk + 16 + koff) * LL, 0, 1);
#pragma unroll
    for (int u = 0; u < 4; ++u) {
      int k = kk + u * 4;
      int k0 = k + koff;
      v2f bv; bv.x = bcol[(size_t)k0 * LL]; bv.y = bcol[(size_t)(k0 + 1) * LL];
      v2f a0; a0.x = arow0[k]; a0.y = arow0[k + 1];
      v2f a1; a1.x = arow1[k]; a1.y = arow1[k + 1];
      v2f a2; a2.x = arow2[k]; a2.y = arow2[k + 1];
      v2f a3; a3.x = arow3[k]; a3.y = arow3[k + 1];
      acc0 = wmma4(a0, bv, acc0);
      acc1 = wmma4(a1, bv, acc1);
      acc2 = wmma4(a2, bv, acc2);
      acc3 = wmma4(a3, bv, acc3);
    }
  }
  { // K tail: k = 960 (961..963 zero-padded)
    v2f bv = {0.0f, 0.0f};
    v2f a0 = {0.0f, 0.0f}, a1 = {0.0f, 0.0f}, a2 = {0.0f, 0.0f}, a3 = {0.0f, 0.0f};
    if (half == 0) {                              // koff==0 lanes hold K=960
      bv.x = bcol[(size_t)960 * LL];
      a0.x = arow0[960]; a1.x = arow1[960];
      a2.x = arow2[960]; a3.x = arow3[960];
    }
    acc0 = wmma4(a0, bv, acc0);
    acc1 = wmma4(a1, bv, acc1);
    acc2 = wmma4(a2, bv, acc2);
    acc3 = wmma4(a3, bv, acc3);
  }
  if (mcol < LL) {
    float* Tib = Ti + (size_t)b * LL * LL + mcol;
#pragma unroll
    for (int r = 0; r < 8; ++r) {
      int row = tn * 64 + half * 8 + r;
      if (row      < LL) Tib[(size_t)(row     ) * LL] = acc0[r] + mcb[row]      + mfb[row];
      if (row + 16 < LL) Tib[(size_t)(row + 16) * LL] = acc1[r] + mcb[row + 16] + mfb[row + 16];
      if (row + 32 < LL) Tib[(size_t)(row + 32) * LL] = acc2[r] + mcb[row + 32] + mfb[row + 32];
      if (row + 48 < LL) Tib[(size_t)(row + 48) * LL] = acc3[r] + mcb[row + 48] + mfb[row + 48];
    }
  }
}

// ---------------- masked softmax row kernel: Am to d_out ----------------
__global__ void k_mask_softmax(const float* __restrict__ A,
                               const float* __restrict__ Ti,
                               float* __restrict__ Am) {
  __shared__ float s[LL];
  __shared__ float mb[LL];
  __shared__ float red[256];
  int l = blockIdx.x, b = blockIdx.y, t = threadIdx.x;
  const float* Ar = A  + ((size_t)b * LL + l) * LL;
  const float* Tr = Ti + ((size_t)b * LL + l) * LL;
  const float scale = 1.0f / 14.0f;   // FOUT^-0.5 = 1/sqrt(196)
  float lmax = -3.4e38f;
  for (int m = t; m < LL; m += 256) {
    float tv = Tr[m];
    float mask = (tv >= 0.0f) ? (1.0f / (1.0f + expf(-tv)))
                              : (expf(tv) / (1.0f + expf(tv)));
    float sv = Ar[m] * mask * scale;
    s[m]  = sv;
    mb[m] = (mask != 0.0f) ? 1.0f : 0.0f;
    lmax = fmaxf(lmax, sv);
  }
  red[t] = lmax; __syncthreads();
  for (int o = 128; o > 0; o >>= 1) {
    if (t < o) red[t] = fmaxf(red[t], red[t + o]);
    __syncthreads();
  }
  float rmax = red[0]; __syncthreads();
  float lsum = 0.0f;
  for (int m = t; m < LL; m += 256) { float e = expf(s[m] - rmax); s[m] = e; lsum += e; }
  red[t] = lsum; __syncthreads();
  for (int o = 128; o > 0; o >>= 1) {
    if (t < o) red[t] += red[t + o];
    __syncthreads();
  }
  float inv = 1.0f / red[0];
  float* Or = Am + ((size_t)b * LL + l) * LL;
  for (int m = t; m < LL; m += 256) Or[m] = s[m] * inv * mb[m];
}

// ---------------- WMMA GEMM: y[b] = Am[b](961x961) @ p62[b](961x1568) -------
__global__ void k_gemm_y(const float* __restrict__ Am,
                         const float* __restrict__ p62,
                         float* __restrict__ y) {
  int tl = blockIdx.x, b = blockIdx.z;
  int wave = threadIdx.x >> 5, lane = threadIdx.x & 31;
  int tf = blockIdx.y * 4 + wave;                 // 100 col tiles (98 valid)
  int half = lane >> 4, i16 = lane & 15, koff = half * 2;
  const float* Amb = Am + (size_t)b * LL * LL;
  const float* arow0 = Amb + (size_t)clampi(tl * 64 +  0 + i16, LL - 1) * LL + koff;
  const float* arow1 = Amb + (size_t)clampi(tl * 64 + 16 + i16, LL - 1) * LL + koff;
  const float* arow2 = Amb + (size_t)clampi(tl * 64 + 32 + i16, LL - 1) * LL + koff;
  const float* arow3 = Amb + (size_t)clampi(tl * 64 + 48 + i16, LL - 1) * LL + koff;
  int fcol = tf * 16 + i16;
  const float* bcol = p62 + (size_t)b * LL * F62 + clampi(fcol, F62 - 1);
  v8f acc0 = {}, acc1 = {}, acc2 = {}, acc3 = {};
#pragma clang loop unroll(disable)
  for (int kk = 0; kk < 960; kk += 16) {
    __builtin_prefetch(bcol + (size_t)(kk + 16 + koff) * F62, 0, 1);
#pragma unroll
    for (int u = 0; u < 4; ++u) {
      int k = kk + u * 4;
      int k0 = k + koff;
      v2f bv; bv.x = bcol[(size_t)k0 * F62]; bv.y = bcol[(size_t)(k0 + 1) * F62];
      v2f a0; a0.x = arow0[k]; a0.y = arow0[k + 1];
      v2f a1; a1.x = arow1[k]; a1.y = arow1[k + 1];
      v2f a2; a2.x = arow2[k]; a2.y = arow2[k + 1];
      v2f a3; a3.x = arow3[k]; a3.y = arow3[k + 1];
      acc0 = wmma4(a0, bv, acc0);
      acc1 = wmma4(a1, bv, acc1);
      acc2 = wmma4(a2, bv, acc2);
      acc3 = wmma4(a3, bv, acc3);
    }
  }
  { // K tail: k = 960
    v2f bv = {0.0f, 0.0f};
    v2f a0 = {0.0f, 0.0f}, a1 = {0.0f, 0.0f}, a2 = {0.0f, 0.0f}, a3 = {0.0f, 0.0f};
    if (half == 0) {
      bv.x = bcol[(size_t)960 * F62];
      a0.x = arow0[960]; a1.x = arow1[960];
      a2.x = arow2[960]; a3.x = arow3[960];
    }
    acc0 = wmma4(a0, bv, acc0);
    acc1 = wmma4(a1, bv, acc1);
    acc2 = wmma4(a2, bv, acc2);
    acc3 = wmma4(a3, bv, acc3);
  }
  if (fcol < F62) {
    float* yb = y + (size_t)b * LL * F62 + fcol;
#pragma unroll
    for (int r = 0; r < 8; ++r) {
      int row = tl * 64 + half * 8 + r;
      if (row      < LL) yb[(size_t)(row     ) * F62] = acc0[r];
      if (row + 16 < LL) yb[(size_t)(row + 16) * F62] = acc1[r];
      if (row + 32 < LL) yb[(size_t)(row + 32) * F62] = acc2[r];
      if (row + 48 < LL) yb[(size_t)(row + 48) * F62] = acc3[r];
    }
  }
}

// ---------------- fold (gather form, no atomics) ----------------
__global__ void k_fold(const float* __restrict__ y,   // (B,LL,F62)
                       float* __restrict__ out) {     // (B,CIN,128,128)
  int idx = blockIdx.x * blockDim.x + threadIdx.x;
  int total = BB * CIN * HW;
  if (idx >= total) return;
  int wq = idx & 127;
  int hq = (idx >> 7) & 127;
  int c  = (idx >> 14) & 31;
  int b  = idx >> 19;
  int lh_lo = (hq > 6) ? ((hq - 3) >> 2) : 0;
  int lh_hi = (hq >> 2); if (lh_hi > 30) lh_hi = 30;
  int lw_lo = (wq > 6) ? ((wq - 3) >> 2) : 0;
  int lw_hi = (wq >> 2); if (lw_hi > 30) lw_hi = 30;
  float acc = 0.0f;
  for (int lh = lh_lo; lh <= lh_hi; ++lh) {
    int kh = hq - lh * STR;
    for (int lw = lw_lo; lw <= lw_hi; ++lw) {
      int kw = wq - lw * STR;
      acc += y[((size_t)b * LL + (lh * NHT + lw)) * F62 + c * 49 + kh * 7 + kw];
    }
  }
  out[idx] = acc;
}

// ---------------- launch ----------------
extern "C" void kernel_launch(void* const* d_in, const int* in_sizes, int n_in,
                              void* d_out, int out_size, void* d_ws, size_t ws_size,
                              hipStream_t stream) {
  const float* x       = (const float*)d_in[0];
  const float* g_w     = (const float*)d_in[1];
  const float* g_b     = (const float*)d_in[2];
  const float* w_w     = (const float*)d_in[3];
  const float* w_b     = (const float*)d_in[4];
  const float* th_w    = (const float*)d_in[5];
  const float* th_b    = (const float*)d_in[6];
  const float* fc1_w   = (const float*)d_in[7];
  const float* fc1_b   = (const float*)d_in[8];
  const float* mconv_w = (const float*)d_in[9];
  const float* mconv_b = (const float*)d_in[10];
  const float* mfc_w   = (const float*)d_in[11];
  const float* mfc_b   = (const float*)d_in[12];
  const float* c1_w    = (const float*)d_in[13];
  const float* c2_w    = (const float*)d_in[14];

  float* Am_out    = (float*)d_out;                         // (B,LL,LL)
  float* final_out = Am_out + (size_t)BB * LL * LL;         // (B,CIN,128,128)

  // workspace layout (floats), with reuse; ~157 MB total
  float* ws   = (float*)d_ws;
  float* x1   = ws;                                  // 2,097,152
  float* x2   = x1  + (size_t)BB * CI  * HW;         // 2,097,152
  float* x3   = x2  + (size_t)BB * CI  * HW;         // 4,194,304
  float* wf   = x3  + (size_t)BB * CIN * HW;         // 1,506,848
  float* xf   = wf  + (size_t)BB * LL * FOUT;        // 1,506,848
  float* p62  = xf  + (size_t)BB * LL * FOUT;        // 12,054,784
  float* A    = p62 + (size_t)BB * LL * F62;         // 7,388,168
  float* Ti   = A   + (size_t)BB * LL * LL;          // 7,388,168
  float* Mw   = Ti  + (size_t)BB * LL * LL;          //   923,521
  float* yb   = A;        // reuse A+Ti region (12.05M <= 14.78M floats)
  float* fout = x1;       // reuse x1+x2 region (exactly 4,194,304)
  float* o1   = wf;       // reuse wf+xf region (2,097,152 <= 3,013,696)

  const int TPB = 256;

  // 1) x1, x2 : 3x3 reflect convs (32 -> 16)
  {
    int total = BB * CI * HW;
    dim3 g((total + TPB - 1) / TPB);
    k_conv3x3_reflect<<<g, TPB, 0, stream>>>(x, g_w, g_b, x1, CIN, CI, 0);
    k_conv3x3_reflect<<<g, TPB, 0, stream>>>(x, w_w, w_b, x2, CIN, CI, 0);
  }
  // 2) x3 = 1x1 conv on x1 (16 -> 32)
  {
    int total = BB * CIN * HW;
    dim3 g((total + TPB - 1) / TPB);
    k_conv1x1<<<g, TPB, 0, stream>>>(x1, th_w, th_b, nullptr, x3, CI, CIN, 0);
  }
  // 3) wf (from x2), xf (from x1)
  {
    int total = BB * LL * FOUT;
    dim3 g((total + TPB - 1) / TPB);
    k_patch_fc<<<g, TPB, 0, stream>>>(x2, fc1_w, fc1_b, wf);
    k_patch_fc<<<g, TPB, 0, stream>>>(x1, fc1_w, fc1_b, xf);
  }
  // 4) p62 and Mw
  {
    int total = BB * LL * F62;
    dim3 g((total + TPB - 1) / TPB);
    k_unfold62<<<g, TPB, 0, stream>>>(x3, p62);
    k_addmw<<<(LL * LL + TPB - 1) / TPB, TPB, 0, stream>>>(mconv_w, mfc_w, Mw);
  }
  // 5) A = wf @ xf^T  (WMMA f32; 4 waves/block, 64x16 per wave)
  k_gemm_score<<<dim3(16, 16, BB), 128, 0, stream>>>(wf, xf, A);
  // 6) Ti = Mw @ A + bias  (WMMA f32)
  k_gemm_ti<<<dim3(16, 16, BB), 128, 0, stream>>>(Mw, mconv_b, mfc_b, A, Ti);
  // 7) Am = softmax(A*sigmoid(Ti)*scale) * (sigmoid(Ti)!=0)  -> d_out
  k_mask_softmax<<<dim3(LL, BB), 256, 0, stream>>>(A, Ti, Am_out);
  // 8) y = Am @ p62  (WMMA f32) -> reuses A/Ti region
  k_gemm_y<<<dim3(16, 25, BB), 128, 0, stream>>>(Am_out, p62, yb);
  // 9) fold y -> (B,32,128,128)
  {
    int total = BB * CIN * HW;
    dim3 g((total + TPB - 1) / TPB);
    k_fold<<<g, TPB, 0, stream>>>(yb, fout);
  }
  // 10) o1 = leaky(conv3x3_reflect(fout, c1_w)) (32 -> 16, no bias)
  {
    int total = BB * CI * HW;
    dim3 g((total + TPB - 1) / TPB);
    k_conv3x3_reflect<<<g, TPB, 0, stream>>>(fout, c1_w, nullptr, o1, CIN, CI, 1);
  }
  // 11) final = x + leaky(conv1x1(o1, c2_w)) (16 -> 32, no bias)
  {
    int total = BB * CIN * HW;
    dim3 g((total + TPB - 1) / TPB);
    k_conv1x1<<<g, TPB, 0, stream>>>(o1, c2_w, nullptr, x, final_out, CI, CIN, 1);
  }
}